// GCN_62586263437733
// MI455X (gfx1250) — compile-verified
//
#include <hip/hip_runtime.h>
#include <math.h>

typedef __attribute__((ext_vector_type(2))) float v2f;
typedef __attribute__((ext_vector_type(8))) float v8f;

#define NFEAT 512
#define NHID  64
#define NCLASS 40
#define LDS_STRIDE 516   // 512 + 4 pad: (col*516+k)%64 spreads 16 cols over distinct banks

// ---------------------------------------------------------------------------
// Transpose: dst[c*R + r] = src[r*C + c]   (tiny one-time cost for W1, W2)
// ---------------------------------------------------------------------------
__global__ void gcn_transpose(const float* __restrict__ src, float* __restrict__ dst,
                              int R, int C) {
  const int i = blockIdx.x * blockDim.x + threadIdx.x;
  if (i >= R * C) return;
  const int r = i / C, c = i % C;
  dst[(long long)c * R + r] = src[i];
}

// ---------------------------------------------------------------------------
// GEMM1: fp1 = x @ W1   ([N,512] x [512,64] -> [N,64]), fp32 WMMA 16x16x4.
// W1 comes in pre-transposed (W1t[64][512]) and is staged into padded LDS so
// each B fragment is ONE conflict-free ds b64 load; vmem is dedicated to the
// HBM stream of x. One wave = one 16x64 tile (4 v8f accumulators).
// ---------------------------------------------------------------------------
__global__ __launch_bounds__(256) void gcn_gemm1_wmma(
    const float* __restrict__ X, const float* __restrict__ W1t,
    float* __restrict__ FP1, int nrows) {
  extern __shared__ float sW[];  // [64][LDS_STRIDE]

  // Cooperative stage: 64*512 floats as float4 (16B-aligned in both spaces)
  for (int idx = threadIdx.x; idx < 64 * 128; idx += 256) {
    const int col = idx >> 7;            // 0..63
    const int kk  = (idx & 127) << 2;    // 0,4,...,508
    *(float4*)(sW + col * LDS_STRIDE + kk) =
        *(const float4*)(W1t + ((long long)col << 9) + kk);
  }
  __syncthreads();

  const int lane = threadIdx.x & 31;
  const int wave = threadIdx.x >> 5;
  const int tile = blockIdx.x * 8 + wave;   // wave-uniform
  const int m0 = tile * 16;
  if (m0 >= nrows) return;                  // uniform per wave: EXEC stays all-ones

  const int half = lane >> 4;               // 0: K={k,k+1}, 1: K={k+2,k+3}
  const int l16  = lane & 15;
  int row = m0 + l16;
  if (row >= nrows) row = nrows - 1;        // clamp loads; stores guarded
  const float* __restrict__ xrow = X + (long long)row * NFEAT;

  const float* __restrict__ s0 = sW + (l16 +  0) * LDS_STRIDE;
  const float* __restrict__ s1 = sW + (l16 + 16) * LDS_STRIDE;
  const float* __restrict__ s2 = sW + (l16 + 32) * LDS_STRIDE;
  const float* __restrict__ s3 = sW + (l16 + 48) * LDS_STRIDE;

  v8f acc0 = {}, acc1 = {}, acc2 = {}, acc3 = {};

  for (int k0 = 0; k0 < NFEAT; k0 += 4) {
    const int ka = k0 + half * 2;
    v2f a;
    a.x = xrow[ka];
    a.y = xrow[ka + 1];

    const v2f b0 = *(const v2f*)(s0 + ka);
    const v2f b1 = *(const v2f*)(s1 + ka);
    const v2f b2 = *(const v2f*)(s2 + ka);
    const v2f b3 = *(const v2f*)(s3 + ka);

    acc0 = __builtin_amdgcn_wmma_f32_16x16x4_f32(false, a, false, b0, (short)0, acc0, false, false);
    acc1 = __builtin_amdgcn_wmma_f32_16x16x4_f32(false, a, false, b1, (short)0, acc1, false, false);
    acc2 = __builtin_amdgcn_wmma_f32_16x16x4_f32(false, a, false, b2, (short)0, acc2, false, false);
    acc3 = __builtin_amdgcn_wmma_f32_16x16x4_f32(false, a, false, b3, (short)0, acc3, false, false);
  }

  // C/D layout: VGPR r -> M = m0 + r + 8*half, N = l16
  for (int r = 0; r < 8; ++r) {
    const int m = m0 + r + half * 8;
    if (m < nrows) {
      float* __restrict__ o = FP1 + (long long)m * NHID + l16;
      o[0]  = acc0[r];
      o[16] = acc1[r];
      o[32] = acc2[r];
      o[48] = acc3[r];
    }
  }
}

// ---------------------------------------------------------------------------
// GEMM2: fp2 = relu(u1_raw + b1) @ W2  ([N,64] x [64,40] -> [N,40]).
// W2 pre-transposed (W2t[40][64], 10 KB: WGP$-resident) -> b64 B loads.
// Bias + ReLU fused into the A-fragment load. 3 column tiles (last guarded).
// ---------------------------------------------------------------------------
__global__ __launch_bounds__(256) void gcn_gemm2_wmma(
    const float* __restrict__ U1, const float* __restrict__ B1,
    const float* __restrict__ W2t, float* __restrict__ FP2, int nrows) {
  const int lane = threadIdx.x & 31;
  const int wave = threadIdx.x >> 5;
  const int tile = blockIdx.x * 8 + wave;
  const int m0 = tile * 16;
  if (m0 >= nrows) return;

  const int half = lane >> 4;
  const int l16  = lane & 15;
  int row = m0 + l16;
  if (row >= nrows) row = nrows - 1;
  const float* __restrict__ urow = U1 + (long long)row * NHID;

  int c2 = l16 + 32; if (c2 >= NCLASS) c2 = NCLASS - 1;  // clamp load, store guarded
  const float* __restrict__ w0 = W2t + (long long)(l16 +  0) * NHID;
  const float* __restrict__ w1 = W2t + (long long)(l16 + 16) * NHID;
  const float* __restrict__ w2 = W2t + (long long)c2 * NHID;

  v8f acc0 = {}, acc1 = {}, acc2 = {};

  for (int k0 = 0; k0 < NHID; k0 += 4) {
    const int ka = k0 + half * 2;
    const v2f u  = *(const v2f*)(urow + ka);
    const v2f bb = *(const v2f*)(B1 + ka);
    v2f a;
    a.x = fmaxf(u.x + bb.x, 0.0f);
    a.y = fmaxf(u.y + bb.y, 0.0f);

    const v2f b0 = *(const v2f*)(w0 + ka);
    const v2f b1 = *(const v2f*)(w1 + ka);
    const v2f b2 = *(const v2f*)(w2 + ka);

    acc0 = __builtin_amdgcn_wmma_f32_16x16x4_f32(false, a, false, b0, (short)0, acc0, false, false);
    acc1 = __builtin_amdgcn_wmma_f32_16x16x4_f32(false, a, false, b1, (short)0, acc1, false, false);
    acc2 = __builtin_amdgcn_wmma_f32_16x16x4_f32(false, a, false, b2, (short)0, acc2, false, false);
  }

  for (int r = 0; r < 8; ++r) {
    const int m = m0 + r + half * 8;
    if (m < nrows) {
      float* __restrict__ o = FP2 + (long long)m * NCLASS;
      o[l16]      = acc0[r];
      o[l16 + 16] = acc1[r];
      if (l16 + 32 < NCLASS) o[l16 + 32] = acc2[r];
    }
  }
}

// ---------------------------------------------------------------------------
// SpMM: out[dst] += feats[src] * w. Each wave grabs a batch of 32 edges with
// COALESCED per-lane metadata loads, then broadcasts via __shfl; lanes stride
// the feature dim with fp32 global atomics (L2-resident gathers + atomics).
// ---------------------------------------------------------------------------
__global__ __launch_bounds__(256) void gcn_spmm_atomic(
    const long long* __restrict__ src, const long long* __restrict__ dst,
    const float* __restrict__ ew, const float* __restrict__ feats,
    float* __restrict__ out, int nedges, int D) {
  const int lane  = threadIdx.x & 31;
  const int gwave = (blockIdx.x * blockDim.x + threadIdx.x) >> 5;
  const int nwave = (gridDim.x * blockDim.x) >> 5;

  for (int base = gwave * 32; base < nedges; base += nwave * 32) {
    const int e = base + lane;
    long long s = 0, d = 0;
    float w = 0.0f;
    if (e < nedges) { s = src[e]; d = dst[e]; w = ew[e]; }
    const int cnt = (nedges - base < 32) ? (nedges - base) : 32;

    for (int j = 0; j < cnt; ++j) {
      const long long sj = __shfl(s, j);
      const long long dj = __shfl(d, j);
      const float     wj = __shfl(w, j);
      const float* __restrict__ fs = feats + sj * D;
      float* __restrict__ od = out + dj * D;
      for (int f = lane; f < D; f += 32) {
        atomicAdd(&od[f], fs[f] * wj);
      }
    }
  }
}

// ---------------------------------------------------------------------------
// Zero-fill (graph-capture-safe, deterministic)
// ---------------------------------------------------------------------------
__global__ void gcn_zero(float* __restrict__ p, long long n) {
  const long long i = (long long)blockIdx.x * blockDim.x + threadIdx.x;
  if (i < n) p[i] = 0.0f;
}

// ---------------------------------------------------------------------------
// Finalize: u2 = u2_acc + b2 (in place), res = log_softmax(u2). One thread/row.
// ---------------------------------------------------------------------------
__global__ __launch_bounds__(256) void gcn_logsoftmax(
    float* __restrict__ u2, const float* __restrict__ b2,
    float* __restrict__ res, int nrows) {
  const int i = blockIdx.x * blockDim.x + threadIdx.x;
  if (i >= nrows) return;
  float v[NCLASS];
  float mx = -INFINITY;
  float* __restrict__ urow = u2 + (long long)i * NCLASS;
#pragma unroll
  for (int c = 0; c < NCLASS; ++c) {
    const float t = urow[c] + b2[c];
    v[c] = t;
    mx = fmaxf(mx, t);
  }
  float s = 0.0f;
#pragma unroll
  for (int c = 0; c < NCLASS; ++c) s += expf(v[c] - mx);
  const float lse = logf(s) + mx;
  float* __restrict__ rrow = res + (long long)i * NCLASS;
#pragma unroll
  for (int c = 0; c < NCLASS; ++c) {
    urow[c] = v[c];
    rrow[c] = v[c] - lse;
  }
}

// ---------------------------------------------------------------------------
// Host launcher
// d_in: x, W1, b1, W2, b2, edge_index(int64,[2,E]), edge_weight
// d_out: res [N*40] | fp1 [N*64] | u2 [N*40]
// d_ws:  u1_acc [N*64] | fp2 [N*40] | W1t [64*512] | W2t [40*64]
// ---------------------------------------------------------------------------
extern "C" void kernel_launch(void* const* d_in, const int* in_sizes, int n_in,
                              void* d_out, int out_size, void* d_ws, size_t ws_size,
                              hipStream_t stream) {
  (void)n_in; (void)out_size; (void)ws_size;

  const float*     x  = (const float*)d_in[0];
  const float*     W1 = (const float*)d_in[1];
  const float*     b1 = (const float*)d_in[2];
  const float*     W2 = (const float*)d_in[3];
  const float*     b2 = (const float*)d_in[4];
  const long long* ei = (const long long*)d_in[5];
  const float*     ew = (const float*)d_in[6];

  const int N = in_sizes[0] / NFEAT;   // 100000
  const int E = in_sizes[6];           // 1600000
  const long long Nll = N;

  float* res = (float*)d_out;                       // [N, 40]
  float* fp1 = res + Nll * NCLASS;                  // [N, 64]
  float* u2  = fp1 + Nll * NHID;                    // [N, 40]

  float* u1acc = (float*)d_ws;                      // [N, 64]
  float* fp2   = u1acc + Nll * NHID;                // [N, 40]
  float* W1t   = fp2 + Nll * NCLASS;                // [64, 512]
  float* W2t   = W1t + NFEAT * NHID;                // [40, 64]

  const long long* src = ei;                        // edge_index[0]
  const long long* dstp = ei + E;                   // edge_index[1]

  const int tiles      = (N + 15) / 16;
  const int gemmBlocks = (tiles + 7) / 8;           // 8 waves (one tile each) per block
  const size_t ldsBytes = (size_t)NHID * LDS_STRIDE * sizeof(float);  // 132096

  // 0) one-time weight transposes (contiguous-k B fragments)
  gcn_transpose<<<(NFEAT * NHID + 255) / 256, 256, 0, stream>>>(W1, W1t, NFEAT, NHID);
  gcn_transpose<<<(NHID * NCLASS + 255) / 256, 256, 0, stream>>>(W2, W2t, NHID, NCLASS);

  // 1) fp1 = x @ W1  (direct into its d_out slot; also the SpMM1 source)
  gcn_gemm1_wmma<<<gemmBlocks, 256, ldsBytes, stream>>>(x, W1t, fp1, N);

  // 2) u1_acc = 0; u1_acc += scatter(fp1[src] * w -> dst)
  {
    const long long n = Nll * NHID;
    gcn_zero<<<(int)((n + 255) / 256), 256, 0, stream>>>(u1acc, n);
  }
  gcn_spmm_atomic<<<8192, 256, 0, stream>>>(src, dstp, ew, fp1, u1acc, E, NHID);

  // 3) fp2 = relu(u1_acc + b1) @ W2
  gcn_gemm2_wmma<<<gemmBlocks, 256, 0, stream>>>(u1acc, b1, W2t, fp2, N);

  // 4) u2 = 0; u2 += scatter(fp2[src] * w -> dst)   (accumulated in d_out slot)
  {
    const long long n = Nll * NCLASS;
    gcn_zero<<<(int)((n + 255) / 256), 256, 0, stream>>>(u2, n);
  }
  gcn_spmm_atomic<<<8192, 256, 0, stream>>>(src, dstp, ew, fp2, u2, E, NCLASS);

  // 5) u2 += b2 (in place), res = log_softmax(u2)
  gcn_logsoftmax<<<(N + 255) / 256, 256, 0, stream>>>(u2, b2, res, N);
}